// GraphEmbedder_77807627534826
// MI455X (gfx1250) — compile-verified
//
#include <hip/hip_runtime.h>

typedef __attribute__((ext_vector_type(2))) float v2f;
typedef __attribute__((ext_vector_type(4))) float v4f;
typedef __attribute__((ext_vector_type(8))) float v8f;

#define NN      1024
#define IN_DIM  32
#define S_DIM   64
#define P_DIM   32
#define SM_STRIDE 66   // pad 64 -> 66: keeps 8B alignment, kills bank conflicts

// ---------------------------------------------------------------------------
// Kernel 1: one wave per 16-row M-tile.
//   single = x @ Ws^T + b          (WMMA f32 16x16x4, K=32 -> 8 steps, 4 N-tiles)
//   p      = 0.5 * single @ Wp^T   (WMMA f32 16x16x4, K=64 -> 16 steps, 2 N-tiles)
// single tile is staged through LDS to re-shape C/D layout -> A layout.
// ---------------------------------------------------------------------------
__global__ __launch_bounds__(32) void ge_compute(
    const float* __restrict__ x,       // [NN, IN_DIM]
    const float* __restrict__ Ws,      // [S_DIM, IN_DIM]
    const float* __restrict__ bs,      // [S_DIM]
    const float* __restrict__ Wp,      // [P_DIM, S_DIM]
    float* __restrict__ out_single,    // [NN, S_DIM]
    float* __restrict__ p_ws)          // [NN, P_DIM]
{
    __shared__ float sm[16 * SM_STRIDE];

    const int lane = threadIdx.x;      // wave32
    const int m    = lane & 15;        // row within tile (A/B/N index)
    const int kh   = lane >> 4;        // K-half selector (0/1)
    const int n0   = blockIdx.x * 16;  // global row base

    // ---- GEMM 1: single = x @ Ws^T ----
    v8f acc[4] = {};                   // 4 N-tiles of 16 (S_DIM = 64)
    #pragma unroll
    for (int kk = 0; kk < IN_DIM / 4; ++kk) {
        const int kb = kk * 4 + kh * 2;              // K = kb, kb+1 for this half
        const v2f a = *(const v2f*)&x[(size_t)(n0 + m) * IN_DIM + kb];
        #pragma unroll
        for (int nt = 0; nt < 4; ++nt) {
            // B[k][s] = Ws[s][k]; lane holds s = nt*16 + m
            const v2f b = *(const v2f*)&Ws[(size_t)(nt * 16 + m) * IN_DIM + kb];
            acc[nt] = __builtin_amdgcn_wmma_f32_16x16x4_f32(
                false, a, false, b, (short)0, acc[nt], false, false);
        }
    }

    // ---- bias, store single, stage tile in LDS (row-major [16][S_DIM]) ----
    #pragma unroll
    for (int nt = 0; nt < 4; ++nt) {
        const float bias = bs[nt * 16 + m];
        #pragma unroll
        for (int r = 0; r < 8; ++r) {
            const int row = r + kh * 8;              // C/D layout: M = r (+8 hi half)
            const float v = acc[nt][r] + bias;
            out_single[(size_t)(n0 + row) * S_DIM + nt * 16 + m] = v;
            sm[row * SM_STRIDE + nt * 16 + m] = v;
        }
    }
    __syncthreads();

    // ---- GEMM 2: p = 0.5 * single @ Wp^T ----
    v8f pc[2] = {};                    // 2 N-tiles of 16 (P_DIM = 32)
    #pragma unroll
    for (int kk = 0; kk < S_DIM / 4; ++kk) {
        const int kb = kk * 4 + kh * 2;
        const v2f a = *(const v2f*)&sm[m * SM_STRIDE + kb];
        #pragma unroll
        for (int jt = 0; jt < 2; ++jt) {
            // B[s][j] = Wp[j][s]; lane holds j = jt*16 + m
            const v2f b = *(const v2f*)&Wp[(size_t)(jt * 16 + m) * S_DIM + kb];
            pc[jt] = __builtin_amdgcn_wmma_f32_16x16x4_f32(
                false, a, false, b, (short)0, pc[jt], false, false);
        }
    }
    #pragma unroll
    for (int jt = 0; jt < 2; ++jt) {
        #pragma unroll
        for (int r = 0; r < 8; ++r) {
            const int row = r + kh * 8;
            p_ws[(size_t)(n0 + row) * P_DIM + jt * 16 + m] = 0.5f * pc[jt][r];
        }
    }
}

// ---------------------------------------------------------------------------
// Kernel 2: pair[i][j][:] = p[i][:] + p[j][:].  Pure store-bandwidth.
// 256 threads/block: 32 j-values x 8 float4 chunks. p (128 KB) lives in L2;
// the 134 MB output goes out with non-temporal b128 stores.
// ---------------------------------------------------------------------------
__global__ __launch_bounds__(256) void ge_pair(
    const float* __restrict__ p,       // [NN, P_DIM]
    float* __restrict__ pair)          // [NN, NN, P_DIM]
{
    const int t  = threadIdx.x;
    const int q  = t & 7;              // float4 chunk within a 32-float row
    const int jl = t >> 3;             // 0..31
    const int i  = blockIdx.y;
    const int j  = blockIdx.x * 32 + jl;

    const v4f* pv = (const v4f*)p;     // P_DIM = 32 -> 8 float4 per row
    const v4f pi = pv[(size_t)i * 8 + q];
    const v4f pj = pv[(size_t)j * 8 + q];
    const v4f r  = pi + pj;

    v4f* outv = (v4f*)pair;
    const size_t idx = ((size_t)i * NN + j) * 8 + q;   // float4 units
    __builtin_nontemporal_store(r, &outv[idx]);
}

extern "C" void kernel_launch(void* const* d_in, const int* in_sizes, int n_in,
                              void* d_out, int out_size, void* d_ws, size_t ws_size,
                              hipStream_t stream) {
    (void)in_sizes; (void)n_in; (void)out_size; (void)ws_size;

    const float* x  = (const float*)d_in[0];   // target_vector [1024, 32]
    const float* Ws = (const float*)d_in[1];   // W_single      [64, 32]
    const float* bs = (const float*)d_in[2];   // b_single      [64]
    const float* Wp = (const float*)d_in[3];   // W_pair        [32, 64]

    float* out        = (float*)d_out;
    float* out_single = out;                               // [1024, 64]
    float* out_pair   = out + (size_t)NN * S_DIM;          // [1024, 1024, 32]
    float* p_ws       = (float*)d_ws;                      // [1024, 32] scratch

    ge_compute<<<NN / 16, 32, 0, stream>>>(x, Ws, bs, Wp, out_single, p_ws);
    ge_pair<<<dim3(NN / 32, NN), 256, 0, stream>>>(p_ws, out_pair);
}